// LSTM_SentimentAnalysis_60808146977314
// MI455X (gfx1250) — compile-verified
//
#include <hip/hip_runtime.h>

// ---------------------------------------------------------------------------
// Bidirectional 2-layer LSTM for MI455X (gfx1250), wave32 + f32 WMMA +
// async global->LDS staging (ASYNCcnt pipelined double buffering).
//
//  * One persistent kernel per layer: 32 workgroups = 16 per direction.
//    WG j of a direction owns h columns [j*32, j*32+32) and the matching 4
//    gate slices (i/f/g/o) -> a 64x128 gate tile per timestep.
//  * Per step: g = [h_{t-1} ; x_t] @ [W_hh | W_ih]^T via V_WMMA_F32_16X16X4_F32
//    (K = 1024 layer 0 with fused embedding gather, K = 1536 layer 1).
//    A/B tiles staged with GLOBAL_LOAD_ASYNC_TO_LDS_B128, double-buffered so
//    stage s+1 DMA overlaps stage s WMMAs.
//  * h broadcast through L2 (double-buffered), per-step inter-WG barrier with
//    per-wave release/acquire fences (each CU has its own L0 path).
//  * Layer 1 stores only t = T-1 (reference keeps only last logits).
// ---------------------------------------------------------------------------

typedef float v2f __attribute__((ext_vector_type(2)));
typedef float v8f __attribute__((ext_vector_type(8)));
typedef int   v4i __attribute__((ext_vector_type(4)));

typedef __attribute__((address_space(1))) v4i as1_v4i;   // global int4
typedef __attribute__((address_space(3))) v4i as3_v4i;   // LDS int4

#if defined(__AMDGCN__) && __has_builtin(__builtin_amdgcn_global_load_async_to_lds_b128)
#define ASYNC_LOAD_B128(gsrc, ldst)                                          \
    __builtin_amdgcn_global_load_async_to_lds_b128(                          \
        (as1_v4i*)(unsigned long long)(gsrc),                                \
        (as3_v4i*)(unsigned)(unsigned long long)(ldst), 0, 0)
#else
#define ASYNC_LOAD_B128(gsrc, ldst)                                          \
    asm volatile("global_load_async_to_lds_b128 %0, %1, off"                 \
                 :: "v"((unsigned)(unsigned long long)(ldst)),               \
                    "v"((const void*)(gsrc))                                 \
                 : "memory")
#endif

#if defined(__AMDGCN__) && __has_builtin(__builtin_amdgcn_s_wait_asynccnt)
#define WAIT_ASYNC(N) __builtin_amdgcn_s_wait_asynccnt(N)
#else
#define WAIT_ASYNC(N) asm volatile("s_wait_asynccnt %0" :: "i"(N) : "memory")
#endif

#define T_SEQ   512
#define BATCH   64
#define HDIM    512
#define NWG_DIR 16

#define ALDS_SZ (64 * 20)     // one A buffer (floats); 20-dword rows => 16B rows
#define BLDS_SZ (128 * 20)    // one B buffer (floats)
#define GSTRIDE 130           // gate tile row stride (conflict-free)

__device__ __forceinline__ float sigmf(float x) {
    return 1.0f / (1.0f + __expf(-x));
}

// Inter-workgroup barrier among the NWG_DIR workgroups of one direction.
// Every wave performs release/acquire fences (two CUs per WGP => per-wave
// write-out / invalidate), thread 0 does the arrive+spin.
__device__ __forceinline__ void grid_barrier(unsigned* bar, unsigned target) {
    __threadfence();                 // release: flush this wave's stores
    __syncthreads();
    if (threadIdx.x == 0) {
        atomicAdd(bar, 1u);
        while (__hip_atomic_load(bar, __ATOMIC_RELAXED, __HIP_MEMORY_SCOPE_AGENT) < target) {
            __builtin_amdgcn_s_sleep(1);
        }
    }
    __syncthreads();
    __threadfence();                 // acquire: invalidate this wave's L0 path
}

template <int LAYER>
__global__ __launch_bounds__(256)
void lstm_layer_kernel(const int*   __restrict__ x,      // (B, T) tokens
                       const float* __restrict__ h0,     // (B, 4, H)
                       const float* __restrict__ c0,     // (B, 4, H)
                       const float* __restrict__ emb,    // (VOCAB, 512)
                       const float* __restrict__ wih0, const float* __restrict__ whh0,
                       const float* __restrict__ bih0, const float* __restrict__ bhh0,
                       const float* __restrict__ wih1, const float* __restrict__ whh1,
                       const float* __restrict__ bih1, const float* __restrict__ bhh1,
                       float* __restrict__ ys0,          // (B, T, 1024) layer-0 output
                       float* __restrict__ ys_last,      // (B, 1024)   layer-1 @ t=T-1
                       float* __restrict__ h_buf,        // 2 dirs x 2 bufs x (B, H)
                       unsigned* __restrict__ bar,       // 2 counters (per dir)
                       float* __restrict__ hn_out,       // (B, 4, H)
                       float* __restrict__ cn_out)       // (B, 4, H)
{
    constexpr int IN   = (LAYER == 0) ? 512 : 1024;
    constexpr int KTOT = HDIM + IN;          // 1024 / 1536
    constexpr int NSTG = KTOT / 16;          // K staged in chunks of 16

    const int tid  = threadIdx.x;
    const int wid  = tid >> 5;               // wave id 0..7 (wave32)
    const int lane = tid & 31;
    const int dir  = blockIdx.x >> 4;        // 0 = fwd, 1 = bwd
    const int jsl  = blockIdx.x & 15;        // h-column slice [jsl*32, +32)
    const int idx  = LAYER * 2 + dir;        // state index in h_n/c_n

    const float* wih = dir ? wih1 : wih0;    // (2048, IN)
    const float* whh = dir ? whh1 : whh0;    // (2048, 512)
    const float* bih = dir ? bih1 : bih0;
    const float* bhh = dir ? bhh1 : bhh0;
    unsigned* mybar  = bar + dir;

    // LDS: double-buffered staging + gate tile. Total 64000 bytes.
    __shared__ float A_lds[2 * ALDS_SZ];     // [h;x] tiles: 64 rows x 16 k
    __shared__ float B_lds[2 * BLDS_SZ];     // weight tiles: 128 cols x 16 k
    __shared__ float g_lds[64 * GSTRIDE];    // 64 x 128 gate pre-activations

    // Wave tiling of the 64x128 output tile.
    const int m0    = (wid & 3) * 16;
    const int wh    = wid >> 2;
    const int lo    = lane & 15;
    const int khalf = (lane < 16) ? 0 : 2;   // ISA A/B fragment k-split

    // Gate-phase mapping: thread owns column cr for 8 batch rows.
    const int cr  = tid & 31;
    const int col = jsl * 32 + cr;           // global h column

    const float bi = bih[         col] + bhh[         col];
    const float bf = bih[ 512  + col] + bhh[ 512  + col];
    const float bg = bih[1024  + col] + bhh[1024  + col];
    const float bo = bih[1536  + col] + bhh[1536  + col];

    float* hb = h_buf + dir * (2 * BATCH * HDIM);

    float creg[8];
#pragma unroll
    for (int e = 0; e < 8; ++e) {
        const int b = wid * 8 + e;
        creg[e] = c0[(b * 4 + idx) * HDIM + col];
        hb[b * HDIM + col] = h0[(b * 4 + idx) * HDIM + col];  // buf 0
    }
    unsigned phase = 0;
    grid_barrier(mybar, ++phase * NWG_DIR);

    for (int s = 0; s < T_SEQ; ++s) {
        const int t = dir ? (T_SEQ - 1 - s) : s;
        const float* hcur = hb + (s & 1) * (BATCH * HDIM);
        float*       hnxt = hb + ((s + 1) & 1) * (BATCH * HDIM);

        // Issue one stage of async global->LDS loads (3 b128 per thread:
        // 1 for the A tile, 2 for the B tile). ASYNCcnt += 3 per wave.
        auto issue_stage = [&](int st, int buf) {
            const int kbase = st * 16;       // uniform across block
            {   // A tile: rows = batch, cols = k-chunk (h region then x)
                const int mm = tid >> 2;
                const int kk = (tid & 3) * 4;
                const float* src;
                if (kbase < HDIM) {
                    src = hcur + mm * HDIM + kbase + kk;
                } else {
                    const int f = kbase - HDIM + kk;
                    if (LAYER == 0) {
                        src = emb + (long long)x[mm * T_SEQ + t] * 512 + f;
                    } else {
                        src = ys0 + (long long)(mm * T_SEQ + t) * 1024 + f;
                    }
                }
                ASYNC_LOAD_B128(src, &A_lds[buf * ALDS_SZ + mm * 20 + kk]);
            }
            {   // B tile: 128 gate rows (q*512 + jsl*32 + r) x 16 k
                const int nn = tid >> 1;
                const int kk = (tid & 1) * 8;
                const int n  = (nn >> 5) * HDIM + jsl * 32 + (nn & 31);
                const float* src = (kbase < HDIM)
                    ? whh + (long long)n * HDIM + kbase + kk
                    : wih + (long long)n * IN + (kbase - HDIM) + kk;
                float* d = &B_lds[buf * BLDS_SZ + nn * 20 + kk];
                ASYNC_LOAD_B128(src,     d);
                ASYNC_LOAD_B128(src + 4, d + 4);
            }
        };

        v8f acc[4];
        const v8f vz = {0.f, 0.f, 0.f, 0.f, 0.f, 0.f, 0.f, 0.f};
#pragma unroll
        for (int i = 0; i < 4; ++i) acc[i] = vz;

        issue_stage(0, 0);                   // prologue: stage 0 -> buffer 0
        for (int st = 0; st < NSTG; ++st) {
            const int cb = st & 1;
            if (st + 1 < NSTG) {
                issue_stage(st + 1, cb ^ 1); // overlap next stage's DMA
                WAIT_ASYNC(3);               // stage st's 3 ops retired
            } else {
                WAIT_ASYNC(0);
            }
            __syncthreads();                 // all waves' stage-st data landed

            const float* Ab = &A_lds[cb * ALDS_SZ];
            const float* Bb = &B_lds[cb * BLDS_SZ];
            // ---- 16 x V_WMMA_F32_16X16X4_F32 per wave per stage ----
#pragma unroll
            for (int ks = 0; ks < 4; ++ks) {
                const int kb = ks * 4 + khalf;
                v2f a;
                a.x = Ab[(m0 + lo) * 20 + kb];
                a.y = Ab[(m0 + lo) * 20 + kb + 1];
#pragma unroll
                for (int i = 0; i < 4; ++i) {
                    const int ncol = (wh * 4 + i) * 16 + lo;
                    v2f bm;
                    bm.x = Bb[ncol * 20 + kb];
                    bm.y = Bb[ncol * 20 + kb + 1];
                    acc[i] = __builtin_amdgcn_wmma_f32_16x16x4_f32(
                        false, a, false, bm, (short)0, acc[i], false, false);
                }
            }
            __syncthreads();                 // buffer cb free for stage st+2
        }

        // ---- dump accumulators (ISA C layout: vgpr r -> rows r, r+8) ----
        {
            const int mrow = m0 + ((lane < 16) ? 0 : 8);
#pragma unroll
            for (int i = 0; i < 4; ++i) {
                const int N = (wh * 4 + i) * 16 + lo;
#pragma unroll
                for (int r = 0; r < 8; ++r) {
                    g_lds[(mrow + r) * GSTRIDE + N] = acc[i][r];
                }
            }
        }
        __syncthreads();

        // ---- gates: c' = sig(f)*c + sig(i)*tanh(g);  h' = sig(o)*tanh(c') ----
#pragma unroll
        for (int e = 0; e < 8; ++e) {
            const int b = wid * 8 + e;
            const float gi = g_lds[b * GSTRIDE +       cr] + bi;
            const float gf = g_lds[b * GSTRIDE +  32 + cr] + bf;
            const float gg = g_lds[b * GSTRIDE +  64 + cr] + bg;
            const float go = g_lds[b * GSTRIDE +  96 + cr] + bo;
            const float cc = sigmf(gf) * creg[e] + sigmf(gi) * tanhf(gg);
            creg[e] = cc;
            const float hh = sigmf(go) * tanhf(cc);
            hnxt[b * HDIM + col] = hh;
            if (LAYER == 0) {
                ys0[(long long)(b * T_SEQ + t) * 1024 + dir * HDIM + col] = hh;
            } else if (t == T_SEQ - 1) {
                ys_last[b * 1024 + dir * HDIM + col] = hh;
            }
            if (s == T_SEQ - 1) {            // final carry (both directions)
                hn_out[(b * 4 + idx) * HDIM + col] = hh;
                cn_out[(b * 4 + idx) * HDIM + col] = cc;
            }
        }
        grid_barrier(mybar, ++phase * NWG_DIR);
    }
}

// out[b][o] = tanh(ys_last[b] . fc_w[o] + fc_b[o]);  B=64, OUT=3, K=1024
__global__ __launch_bounds__(192)
void fc_tanh_kernel(const float* __restrict__ ys_last,
                    const float* __restrict__ fcw,
                    const float* __restrict__ fcb,
                    float* __restrict__ out)
{
    const int i = threadIdx.x;        // 0..191
    const int b = i / 3, o = i % 3;
    float s = fcb[o];
    const float* yv = ys_last + b * 1024;
    const float* wv = fcw + o * 1024;
    for (int k = 0; k < 1024; ++k) s += yv[k] * wv[k];
    out[b * 3 + o] = tanhf(s);
}

extern "C" void kernel_launch(void* const* d_in, const int* in_sizes, int n_in,
                              void* d_out, int out_size, void* d_ws, size_t ws_size,
                              hipStream_t stream) {
    (void)in_sizes; (void)n_in; (void)out_size; (void)ws_size;
    // setup_inputs() order: x, h0, c0, then params in insertion order:
    // embedding, {w_ih, w_hh, b_ih, b_hh} for (l,d) = (0,0),(0,1),(1,0),(1,1),
    // fc_w, fc_b.
    const int*   x   = (const int*)  d_in[0];
    const float* h0  = (const float*)d_in[1];
    const float* c0  = (const float*)d_in[2];
    const float* emb = (const float*)d_in[3];
    const float* wih[4], *whh[4], *bih[4], *bhh[4];
    for (int i = 0; i < 4; ++i) {
        wih[i] = (const float*)d_in[4 + i * 4 + 0];
        whh[i] = (const float*)d_in[4 + i * 4 + 1];
        bih[i] = (const float*)d_in[4 + i * 4 + 2];
        bhh[i] = (const float*)d_in[4 + i * 4 + 3];
    }
    const float* fcw = (const float*)d_in[20];
    const float* fcb = (const float*)d_in[21];

    float* out    = (float*)d_out;                 // (64, 3)
    float* hn_out = out + 64 * 3;                  // (64, 4, 512)
    float* cn_out = hn_out + 64 * 4 * 512;         // (64, 4, 512)

    // Workspace layout.
    char* ws = (char*)d_ws;
    unsigned* bar   = (unsigned*)ws;                       // 4 counters
    float* h_buf    = (float*)(ws + 256);                  // 512 KiB
    float* ys0      = (float*)(ws + (1u << 20));           // 128 MiB
    float* ys_last  = ys0 + (long long)64 * 512 * 1024;    // 256 KiB

    (void)hipMemsetAsync(bar, 0, 16, stream);

    lstm_layer_kernel<0><<<dim3(2 * NWG_DIR), dim3(256), 0, stream>>>(
        x, h0, c0, emb,
        wih[0], whh[0], bih[0], bhh[0],
        wih[1], whh[1], bih[1], bhh[1],
        ys0, ys_last, h_buf, bar + 0, hn_out, cn_out);

    lstm_layer_kernel<1><<<dim3(2 * NWG_DIR), dim3(256), 0, stream>>>(
        x, h0, c0, emb,
        wih[2], whh[2], bih[2], bhh[2],
        wih[3], whh[3], bih[3], bhh[3],
        ys0, ys_last, h_buf, bar + 2, hn_out, cn_out);

    fc_tanh_kernel<<<dim3(1), dim3(192), 0, stream>>>(ys_last, fcw, fcb, out);
}